// MHPLSTMCore_22050362098037
// MI455X (gfx1250) — compile-verified
//
#include <hip/hip_runtime.h>
#include <math.h>

typedef float v2f __attribute__((ext_vector_type(2)));
typedef float v8f __attribute__((ext_vector_type(8)));

#define Bb 4
#define Ss 4096
#define Hh 8
#define DI 128
#define DOo 128
#define HD   (Hh*DI)        // 1024
#define KDIM 256            // 2*DI
#define N3DO 384            // 3*DO
#define BS   (Bb*Ss)        // 16384
#define NC   32
#define CHUNK (Ss/NC)       // 128

__device__ __forceinline__ float sigmoidf_(float x){ return 1.0f/(1.0f + __expf(-x)); }

// block size must be 256; returns (mu, rstd)
__device__ __forceinline__ float2 ln_stats(float sum, float sq, float* red, int t, float invN){
    red[t] = sum; __syncthreads();
    for (int st = 128; st > 0; st >>= 1){ if (t < st) red[t] += red[t+st]; __syncthreads(); }
    float mu = red[0] * invN;
    __syncthreads();
    red[t] = sq; __syncthreads();
    for (int st = 128; st > 0; st >>= 1){ if (t < st) red[t] += red[t+st]; __syncthreads(); }
    float var = red[0] * invN - mu*mu;
    __syncthreads();
    return make_float2(mu, rsqrtf(var + 1e-6f));
}

// ---------------- Stage 1: chunked exclusive cumsum along S ----------------
__global__ void k_cumsum_chunk(const float* __restrict__ x, float* __restrict__ csum,
                               float* __restrict__ chunkSum){
    const int d = threadIdx.x;                 // 0..127
    const int chunk = blockIdx.x;              // 0..31
    const int bh = blockIdx.y;                 // 0..31
    const int b = bh >> 3, h = bh & 7;
    float c = 0.f;
    const int s0 = chunk * CHUNK;
    for (int j = 0; j < CHUNK; ++j){
        const int idx = ((b*Ss + s0 + j)*Hh + h)*DI + d;
        __builtin_prefetch(x + idx + 8*Hh*DI, 0, 0);   // global_prefetch_b8, 32KB ahead
        const float v = x[idx];
        csum[idx] = c;                         // exclusive
        c += v;
    }
    chunkSum[(bh*NC + chunk)*DI + d] = c;
}

__global__ void k_chunk_prefix(float* __restrict__ chunkSum){
    const int d = threadIdx.x;
    const int bh = blockIdx.x;
    float c = 0.f;
    for (int ch = 0; ch < NC; ++ch){
        const int idx = (bh*NC + ch)*DI + d;
        const float t = chunkSum[idx];
        chunkSum[idx] = c;                     // exclusive chunk offsets
        c += t;
    }
}

// ---------------- Stage 2: LN over (H,DI) per (b,s), fuse chunk-offset add ----------------
__global__ void k_ln_csum(float* __restrict__ csum, const float* __restrict__ chunkSum,
                          const float* __restrict__ g, const float* __restrict__ beta){
    __shared__ float red[256];
    const int t = threadIdx.x;
    const int bs = blockIdx.x;
    const int b = bs / Ss, s = bs % Ss;
    const int chunk = s >> 7;
    const int base = bs * HD;
    float vals[4];
    float sum = 0.f, sq = 0.f;
    #pragma unroll
    for (int jj = 0; jj < 4; ++jj){
        const int j = t + jj*256;              // j = h*128 + d
        const int h = j >> 7, d = j & 127;
        const float v = csum[base + j] + chunkSum[((b*Hh + h)*NC + chunk)*DI + d];
        vals[jj] = v; sum += v; sq += v*v;
    }
    const float2 st = ln_stats(sum, sq, red, t, 1.0f/(float)HD);
    #pragma unroll
    for (int jj = 0; jj < 4; ++jj){
        const int j = t + jj*256;
        csum[base + j] = (vals[jj] - st.x) * st.y * g[j] + beta[j];
    }
}

// ---------------- WMMA fp32 GEMM: rows = concat(xin, xin2) per head ----------------
// C tile per wave: 16x64 (4 accumulators). Block = 4 waves sharing a 256x64 B panel in LDS.
// LDS layout: pair-interleaved + rotate-by-32 on odd kp rows -> single conflict-free
// ds_load_b64 per B fragment (even-aligned VGPR pair, no register shuffles).
template<int NDIM>
__global__ void __launch_bounds__(128)
k_gemm(const float* __restrict__ xin, const float* __restrict__ xin2,
       const float* __restrict__ W, const float* __restrict__ bias,
       float* __restrict__ out){
    __shared__ float ldsW[(KDIM/2) * 128];     // 128 kp-rows x 128 floats = 64KB
    const int h  = blockIdx.z;
    const int n0 = blockIdx.y * 64;
    const int t  = threadIdx.x;
    const float* Wh = W + h * KDIM * NDIM;
    for (int idx = t; idx < KDIM*64; idx += 128){
        const int k = idx >> 6, n = idx & 63;
        const int kp = k >> 1, par = k & 1;
        const int sw = ((2*n + ((kp & 1) << 5)) & 127) + par;  // pair-interleave + rotate
        ldsW[kp*128 + sw] = Wh[k*NDIM + n0 + n];
    }
    __syncthreads();
    const int wave = t >> 5, lane = t & 31;
    const int mn = lane & 15, kh = lane >> 4;  // A: row=mn ; B: col=mn ; k-half select
    const int rowbase = (blockIdx.x*4 + wave) * 16;
    const int row = rowbase + mn;              // flattened (b,s) index
    const v2f* ap0 = (const v2f*)(xin  + (row*Hh + h)*DI);
    const v2f* ap1 = (const v2f*)(xin2 + (row*Hh + h)*DI);
    __builtin_prefetch(ap1, 0, 3);             // warm the second-half A rows
    const v2f* Bp = (const v2f*)ldsW;          // kp row = 64 v2f
    // lane-constant swizzled column offsets (v2f units within a kp row)
    int off[4];
    #pragma unroll
    for (int nt = 0; nt < 4; ++nt) off[nt] = ((nt + kh) & 3)*16 + mn;

    v8f acc[4] = {{}, {}, {}, {}};
    // half 0: k in [0,128) from xin        (kp = 2j + kh)
    #pragma unroll 4
    for (int j = 0; j < 32; ++j){
        const int kk = 2*j + kh;
        const v2f a = ap0[kk];
        #pragma unroll
        for (int nt = 0; nt < 4; ++nt){
            const v2f b = Bp[kk*64 + off[nt]];
            acc[nt] = __builtin_amdgcn_wmma_f32_16x16x4_f32(false, a, false, b, (short)0, acc[nt], false, false);
        }
    }
    // half 1: k in [128,256) from xin2     (kp = 64 + 2j + kh)
    #pragma unroll 4
    for (int j = 0; j < 32; ++j){
        const int kk = 2*j + kh;
        const v2f a = ap1[kk];
        #pragma unroll
        for (int nt = 0; nt < 4; ++nt){
            const v2f b = Bp[(64 + kk)*64 + off[nt]];
            acc[nt] = __builtin_amdgcn_wmma_f32_16x16x4_f32(false, a, false, b, (short)0, acc[nt], false, false);
        }
    }
    const int rh = kh * 8;
    #pragma unroll
    for (int nt = 0; nt < 4; ++nt){
        const int n = n0 + nt*16 + mn;
        const float bv = bias[h*NDIM + n];
        #pragma unroll
        for (int r = 0; r < 8; ++r){
            const int rr = rowbase + rh + r;   // C layout: vgpr r = row r (+8 for lanes 16..31)
            out[(rr*Hh + h)*NDIM + n] = acc[nt][r] + bv;
        }
    }
}

// ---------------- Stage 4: LN over (H,3,DO)=3072 + gates, in place ----------------
__global__ void k_ln_gates(float* __restrict__ h3,
                           const float* __restrict__ g, const float* __restrict__ beta){
    __shared__ float red[256];
    const int t = threadIdx.x;
    const int bs = blockIdx.x;
    const int base = bs * (Hh*N3DO);           // 3072 contiguous
    float sum = 0.f, sq = 0.f;
    #pragma unroll
    for (int jj = 0; jj < 12; ++jj){
        const float v = h3[base + t + jj*256];
        sum += v; sq += v*v;
    }
    const float2 st = ln_stats(sum, sq, red, t, 1.0f/3072.0f);
    #pragma unroll
    for (int cc = 0; cc < 4; ++cc){
        const int c = t + cc*256;              // channel = h*128 + d
        const int h = c >> 7, d = c & 127;
        const int off = h*N3DO + d;            // == flat index into (H,3,DO) for slot 0
        const float iraw = h3[base + off];
        const float fraw = h3[base + off + 128];
        const float hraw = h3[base + off + 256];
        const float in_ = (iraw - st.x)*st.y * g[off      ] + beta[off      ];
        const float fn  = (fraw - st.x)*st.y * g[off + 128] + beta[off + 128];
        const float hn  = (hraw - st.x)*st.y * g[off + 256] + beta[off + 256];
        h3[base + off]       = sigmoidf_(fn);                      // slot0 <- f-gate
        h3[base + off + 128] = sigmoidf_(in_) * fmaxf(hn, 0.f);    // slot1 <- i*relu(h)
        // slot2 reused below for running f-product
    }
}

// ---------------- Stage 5: chunked scan  c = f*c + i ----------------
__global__ void k_scan_chunk(float* __restrict__ h3, float* __restrict__ cell,
                             float* __restrict__ carryP, float* __restrict__ carryC){
    const int d = threadIdx.x;
    const int chunk = blockIdx.x;
    const int bh = blockIdx.y;
    const int b = bh >> 3, h = bh & 7;
    float c = 0.f, p = 1.f;
    for (int j = 0; j < CHUNK; ++j){
        const int s = chunk*CHUNK + j;
        const int base = ((b*Ss + s)*Hh + h)*N3DO;
        __builtin_prefetch(h3 + base + 8*Hh*N3DO + d, 0, 0);
        const float f = h3[base + d];
        const float i = h3[base + 128 + d];
        p *= f;
        c = f*c + i;
        cell[((b*Ss + s)*Hh + h)*DOo + d] = c;  // B_s (zero-init local)
        h3[base + 256 + d] = p;                 // A_s (running product incl. f_s)
    }
    carryP[(bh*NC + chunk)*DOo + d] = p;
    carryC[(bh*NC + chunk)*DOo + d] = c;
}

__global__ void k_scan_prefix(const float* __restrict__ carryP, const float* __restrict__ carryC,
                              const float* __restrict__ initcx, float* __restrict__ cstart){
    const int d = threadIdx.x;
    const int bh = blockIdx.x;
    const int h = bh & 7;
    float c = initcx[h*DOo + d];
    for (int ch = 0; ch < NC; ++ch){
        const int idx = (bh*NC + ch)*DOo + d;
        cstart[idx] = c;
        c = carryP[idx]*c + carryC[idx];
    }
}

__global__ void k_scan_fixup(float* __restrict__ cell, const float* __restrict__ h3,
                             const float* __restrict__ cstart){
    const int e = blockIdx.x*256 + threadIdx.x;   // < B*S*H*DO
    const int d = e & 127;
    const int bsh = e >> 7;                       // (b*S+s)*H + h
    const int h = bsh & 7;
    const int s = (bsh >> 3) & (Ss - 1);
    const int b = bsh >> 15;
    const int chunk = s >> 7;
    const float A = h3[bsh*N3DO + 256 + d];
    const float cs = cstart[(((b*Hh + h)*NC) + chunk)*DOo + d];
    cell[e] = A*cs + cell[e];
}

// ---------------- Stage 7: LN over (H,DO) + sigmoid(og)*cell ----------------
__global__ void k_ln_out(const float* __restrict__ og, const float* __restrict__ cell,
                         const float* __restrict__ g, const float* __restrict__ beta,
                         float* __restrict__ out){
    __shared__ float red[256];
    const int t = threadIdx.x;
    const int bs = blockIdx.x;
    const int base = bs * HD;                  // 1024 contiguous
    float vals[4];
    float sum = 0.f, sq = 0.f;
    #pragma unroll
    for (int jj = 0; jj < 4; ++jj){
        const float v = og[base + t + jj*256];
        vals[jj] = v; sum += v; sq += v*v;
    }
    const float2 st = ln_stats(sum, sq, red, t, 1.0f/(float)HD);
    #pragma unroll
    for (int jj = 0; jj < 4; ++jj){
        const int j = t + jj*256;
        const float n = (vals[jj] - st.x)*st.y * g[j] + beta[j];
        const float o = sigmoidf_(n) * cell[base + j];
        __builtin_nontemporal_store(o, &out[base + j]);   // written once, never re-read
    }
}

extern "C" void kernel_launch(void* const* d_in, const int* in_sizes, int n_in,
                              void* d_out, int out_size, void* d_ws, size_t ws_size,
                              hipStream_t stream){
    const float* x       = (const float*)d_in[0];   // (B,S,H,DI)
    const float* W_hid   = (const float*)d_in[1];   // (H,256,384)
    const float* b_hid   = (const float*)d_in[2];   // (H,384)
    const float* g_csum  = (const float*)d_in[3];   // (H,DI)
    const float* be_csum = (const float*)d_in[4];
    const float* g_hid   = (const float*)d_in[5];   // (H,3,DO)
    const float* be_hid  = (const float*)d_in[6];
    const float* W_og    = (const float*)d_in[7];   // (H,256,128)
    const float* b_og    = (const float*)d_in[8];   // (H,128)
    const float* g_og    = (const float*)d_in[9];   // (H,DO)
    const float* be_og   = (const float*)d_in[10];
    const float* initcx  = (const float*)d_in[11];  // (H,DO)
    float* out = (float*)d_out;

    float* ws   = (float*)d_ws;
    float* ws0  = ws;                       // 16,777,216 floats: csum -> cell
    float* ws1  = ws + 16777216;            // 50,331,648 floats: h3 -> gates -> og
    float* chunkSum = ws + 67108864;        // 131,072
    float* carryP   = chunkSum + 131072;    // 131,072
    float* carryC   = carryP   + 131072;    // 131,072
    float* cstart   = carryC   + 131072;    // 131,072

    // 1) exclusive cumsum (chunked scan)
    k_cumsum_chunk<<<dim3(NC, Bb*Hh), DI, 0, stream>>>(x, ws0, chunkSum);
    k_chunk_prefix<<<Bb*Hh, DI, 0, stream>>>(chunkSum);
    // 2) LN over (H,DI), fusing chunk-offset add
    k_ln_csum<<<BS, 256, 0, stream>>>(ws0, chunkSum, g_csum, be_csum);
    // 3) h3 = [x, csum] @ W_hid + b_hid   (WMMA fp32)
    k_gemm<N3DO><<<dim3(BS/64, N3DO/64, Hh), 128, 0, stream>>>(x, ws0, W_hid, b_hid, ws1);
    // 4) LN over (H,3,DO) + gates (in place)
    k_ln_gates<<<BS, 256, 0, stream>>>(ws1, g_hid, be_hid);
    // 5) recurrence c = f*c + i (chunked scan)
    k_scan_chunk<<<dim3(NC, Bb*Hh), DOo, 0, stream>>>(ws1, ws0, carryP, carryC);
    k_scan_prefix<<<Bb*Hh, DOo, 0, stream>>>(carryP, carryC, initcx, cstart);
    k_scan_fixup<<<(Bb*Ss*Hh*DOo)/256, 256, 0, stream>>>(ws0, ws1, cstart);
    // 6) og = [x, cell] @ W_og + b_og   (WMMA fp32) -> compact at ws1 base
    k_gemm<DOo><<<dim3(BS/64, DOo/64, Hh), 128, 0, stream>>>(x, ws0, W_og, b_og, ws1);
    // 7) LN over (H,DO) + sigmoid(og)*cell
    k_ln_out<<<BS, 256, 0, stream>>>(ws1, ws0, g_og, be_og, out);
}